// SpatialExternalMemory_403726926418
// MI455X (gfx1250) — compile-verified
//
#include <hip/hip_runtime.h>
#include <stdint.h>

#if defined(__HIP_DEVICE_COMPILE__)
#if !__has_builtin(__builtin_amdgcn_tensor_load_to_lds)
#error "MISSING __builtin_amdgcn_tensor_load_to_lds"
#endif
#if !__has_builtin(__builtin_amdgcn_tensor_store_from_lds)
#error "MISSING __builtin_amdgcn_tensor_store_from_lds"
#endif
#if !__has_builtin(__builtin_amdgcn_s_wait_tensorcnt)
#error "MISSING __builtin_amdgcn_s_wait_tensorcnt"
#endif
#endif

typedef unsigned int u32x4 __attribute__((ext_vector_type(4)));
typedef int          i32x8 __attribute__((ext_vector_type(8)));
typedef int          i32x4 __attribute__((ext_vector_type(4)));

// Problem constants (reference: N=M=1024, H=128, B=65536, w=2 -> K=5, KK=25).
#define BATCH      65536u
#define HDIM       128u          // floats per row (512 B)
#define KK         25u
#define ROWBYTES   512u
#define WAVE_LDS   (KK * ROWBYTES)        // 12800 B staging per wave
#define WAVES_PB   8u                      // 8 waves (batch points) per block

// Descriptor field constants ------------------------------------------------
// group0.dword0: count=1, gather_mode=1 (bit31), 32-bit indices (bit30)
#define G0_LOAD    0xC0000001u
#define G0_STORE   0x00000001u
// group1.dword0: data_size=2 (4 bytes) at bits [17:16]
#define G1_DSIZE   0x00020000u
// group1.dword1: tensor_dim0[15:0] << 16 (tensor_dim0 = 128)
#define G1_TD0LO   (128u << 16)
// group1.dword3: tile_dim0 (=128) in [31:16] | tensor_dim1[31:16] in [15:0]
#define G1_LOAD_D3  ((128u << 16) | 0x10u)   // tensor_dim1 = 0x100000 (1M rows)
#define G1_STORE_D3 ((128u << 16) | 0x19u)   // tensor_dim1 = 0x190000 (1.6M rows)
// store: stride between output "lines" (k-slices) = 65536 * 128 elements
#define OUT_STRIDE 8388608
// cpol: TH in bits [2:0] (gfx12-family CPol). Loads RT (0) -> exploit 192MB L2
// reuse of the 5x5-window rows; store NT (1) -> 800MB write-once stream must
// not evict the reusable source rows.
#define CPOL_LOAD_RT  0
#define CPOL_STORE_NT 1

// One wave per batch point b. 4 TDM gather-loads (8+8+8+1 indexed rows of
// 512B) stage memory[.,.,:] rows into LDS; one TDM 2D store scatters the
// 25 rows to their strided output positions r = k*65536 + b.
__global__ __launch_bounds__(256) void SpatialExternalMemory_tdm(
    const float* __restrict__ mem,          // (1024,1024,128) f32
    const long long* __restrict__ grid_in,  // (B,2) i64
    float* __restrict__ out)                // flat rows r = k*B + b
{
#if defined(__HIP_DEVICE_COMPILE__)
    __shared__ unsigned char stage[WAVES_PB * WAVE_LDS];   // 102400 B / WGP
    // TDM addresses LDS by byte offset; keep the allocation alive.
    // (Observed codegen: &stage[0] == LDS offset 0, matching the descriptors.)
    asm volatile("" :: "v"(&stage[0]));

    const int      uwid     = __builtin_amdgcn_readfirstlane((int)(threadIdx.x >> 5));
    const unsigned b        = blockIdx.x * WAVES_PB + (unsigned)uwid;
    const unsigned lds_base = (unsigned)uwid * WAVE_LDS;

    // (gx, gy) are wave-uniform: broadcast b128 load + readfirstlane -> SGPRs.
    const long long* gp = grid_in + 2u * (size_t)b;
    const int gx = __builtin_amdgcn_readfirstlane((int)gp[0]);
    const int gy = __builtin_amdgcn_readfirstlane((int)gp[1]);

    // 25 source row indices ix*1024 + iy at k = 5*ki + kj (ref: maximum(., 0) clamp).
    unsigned idx[KK];
#pragma unroll
    for (int ki = 0; ki < 5; ++ki) {
        int ix = gx + ki - 2; ix = ix < 0 ? 0 : ix;
#pragma unroll
        for (int kj = 0; kj < 5; ++kj) {
            int iy = gy + kj - 2; iy = iy < 0 ? 0 : iy;
            idx[ki * 5 + kj] = ((unsigned)ix << 10) | (unsigned)iy;
        }
    }

    const unsigned long long ma = (unsigned long long)(uintptr_t)mem;
    const unsigned ma_lo = (unsigned)ma;
    const unsigned ma_hi = (unsigned)((ma >> 32) & 0x01FFFFFFull) | 0x80000000u; // type=2

    // 5th builtin arg: ISA VADDR4 slot is unused for tensor ops -> zeros.
    const i32x8 zero8 = { 0, 0, 0, 0, 0, 0, 0, 0 };

    // ---- 4 gather-mode TDM loads: rows k = 8d .. 8d+nvalid-1 -> LDS -------
#pragma unroll
    for (int d = 0; d < 4; ++d) {
        const int nvalid = (d == 3) ? 1 : 8;
        const u32x4 g0 = { G0_LOAD, lds_base + (unsigned)d * 4096u, ma_lo, ma_hi };
        const i32x8 g1 = { (int)G1_DSIZE, (int)G1_TD0LO, 0, (int)G1_LOAD_D3,
                           nvalid,        128 /*tensor_dim0_stride*/, 0, 0 };
        i32x4 g2, g3;
        if (d < 3) {
            const int o = d * 8;
            g2 = (i32x4){ (int)idx[o+0], (int)idx[o+1], (int)idx[o+2], (int)idx[o+3] };
            g3 = (i32x4){ (int)idx[o+4], (int)idx[o+5], (int)idx[o+6], (int)idx[o+7] };
        } else {
            const int t = (int)idx[24];          // only index 0 is valid
            g2 = (i32x4){ t, t, t, t };
            g3 = (i32x4){ t, t, t, t };
        }
        __builtin_amdgcn_tensor_load_to_lds(g0, g1, g2, g3, zero8, CPOL_LOAD_RT);
    }

    // LDS RAW: gathered rows must land before the store descriptor reads LDS.
    __builtin_amdgcn_s_wait_tensorcnt(0);

    // ---- one 2D TDM store: 25 x 128 tile, line stride 65536*128 elems -----
    const unsigned long long oa =
        (unsigned long long)(uintptr_t)out + (unsigned long long)b * ROWBYTES;
    const u32x4 s0 = { G0_STORE, lds_base, (unsigned)oa,
                       (unsigned)((oa >> 32) & 0x01FFFFFFull) | 0x80000000u };
    const i32x8 s1 = { (int)G1_DSIZE, (int)G1_TD0LO, 0, (int)G1_STORE_D3,
                       (int)KK,       OUT_STRIDE, 0, 0 };
    const i32x4 s2 = { 1, 0, 0, 0 };   // tensor_dim2=1; tile_dim2/3 = 0 -> unused
    const i32x4 s3 = { 0, 0, 0, 0 };
    __builtin_amdgcn_tensor_store_from_lds(s0, s1, s2, s3, zero8, CPOL_STORE_NT);
    // S_ENDPGM performs an implicit wait-idle, draining TENSORcnt.
#else
    (void)mem; (void)grid_in; (void)out;
#endif
}

extern "C" void kernel_launch(void* const* d_in, const int* in_sizes, int n_in,
                              void* d_out, int out_size, void* d_ws, size_t ws_size,
                              hipStream_t stream) {
    (void)in_sizes; (void)n_in; (void)out_size; (void)d_ws; (void)ws_size;

    const float*     mem     = (const float*)d_in[0];
    const long long* grid_in = (const long long*)d_in[1];
    float*           out     = (float*)d_out;

    const unsigned blocks = BATCH / WAVES_PB;   // 8192 blocks x 8 waves
    SpatialExternalMemory_tdm<<<blocks, 256, 0, stream>>>(mem, grid_in, out);
}